// DCNv3_35622458753539
// MI455X (gfx1250) — compile-verified
//
#include <hip/hip_runtime.h>
#include <math.h>

// ---- problem constants (match reference) ----
#define NB   4
#define HH   64
#define WW   64
#define CCH  128
#define GG   4
#define GCH  32
#define PP   9
#define NPIX (NB * HH * WW)          // 16384 pixels
#define KOFF (GG * PP * 2)           // 72
#define KMSK (GG * PP)               // 36

typedef __attribute__((ext_vector_type(2))) float v2f;
typedef __attribute__((ext_vector_type(8))) float v8f;

// ---------------------------------------------------------------------------
// fp32 WMMA GEMM: D[M,N] = A[M,K] @ B[K,N] + bias[N], K compile-time (128).
// One wave32 computes one 16x16 output tile via V_WMMA_F32_16X16X4_F32.
//
// Fragment layouts (ISA 7.12.2, 32-bit A 16x4 / 32-bit C/D 16x16):
//   A: lane half (lane>=16) picks K-pair {2,3} vs {0,1}; v[0..1] walk K inside
//      the pair -> one aligned float2 per lane per step.
//   B: lane half picks K-half, vgpr index walks K inside half -> two loads
//      strided by N.
//   C/D: vgpr i -> row half*8+i, lane&15 -> col.
//
// Out-of-range columns (N not multiple of 16) are handled by CLAMPING the load
// column to N-1: D[m,n] only depends on lane n's B column, so the clamped
// lanes' garbage never reaches stored columns (store is guarded). This keeps
// the inner loop fully convergent (EXEC all-1s, as WMMA requires) with zero
// saveexec traffic.
// ---------------------------------------------------------------------------
template <int K>
__global__ __launch_bounds__(256) void wmma_gemm_f32_bias(
    const float* __restrict__ A, const float* __restrict__ B,
    const float* __restrict__ bias, float* __restrict__ D,
    const int M, const int N)
{
    const int lane   = threadIdx.x & 31;
    const int tilesN = (N + 15) >> 4;
    const int wave   = blockIdx.x * (blockDim.x >> 5) + (threadIdx.x >> 5);
    const int tileM  = wave / tilesN;
    const int tileN  = wave - tileM * tilesN;
    if (tileM * 16 >= M) return;

    const int  half  = lane >> 4;        // 0: K{0,1} / 1: K{2,3} within each step
    const int  l16   = lane & 15;
    const int  arow  = tileM * 16 + l16; // A row for this lane
    const int  col   = tileN * 16 + l16; // D column for this lane
    const int  colL  = col < N ? col : N - 1;   // clamped load column

    const float* __restrict__ Ap = A + (size_t)arow * K + half * 2;
    const float* __restrict__ Bp = B + (size_t)(half * 2) * N + colL;
    const size_t sB = (size_t)N;         // B row stride in floats

    v8f acc = {};
    #pragma unroll 4
    for (int k = 0; k < K; k += 4) {
        const v2f a = *(const v2f*)Ap;   // 8B-aligned: arow*K + half*2 is even
        v2f b;
        b.x = Bp[0];
        b.y = Bp[sB];
        acc = __builtin_amdgcn_wmma_f32_16x16x4_f32(
            /*neg_a=*/false, a, /*neg_b=*/false, b,
            /*c_mod=*/(short)0, acc, /*reuse_a=*/false, /*reuse_b=*/false);
        Ap += 4;
        Bp += 4 * sB;
    }

    if (col < N) {
        const float bv = bias[col];
        const float* accp = (const float*)&acc;
        float* __restrict__ Dp = D + (size_t)(tileM * 16 + half * 8) * N + col;
        #pragma unroll
        for (int i = 0; i < 8; ++i) {
            Dp[0] = accp[i] + bv;
            Dp += N;
        }
    }
}

// ---------------------------------------------------------------------------
// Depthwise 3x3 conv + bias -> LayerNorm(C, eps=1e-6) -> exact GELU.
// One 128-thread block per pixel; LDS tree reduction for mean / E[x^2].
// ---------------------------------------------------------------------------
__global__ __launch_bounds__(128) void dw_ln_gelu(
    const float* __restrict__ in, const float* __restrict__ dwk,
    const float* __restrict__ dwb, const float* __restrict__ gamma,
    const float* __restrict__ beta, float* __restrict__ x1)
{
    const int pix = blockIdx.x;
    const int c   = threadIdx.x;
    const int x   = pix % WW;
    const int y   = (pix / WW) % HH;
    const int n   = pix / (HH * WW);

    float acc = dwb[c];
    #pragma unroll
    for (int ky = 0; ky < 3; ++ky) {
        #pragma unroll
        for (int kx = 0; kx < 3; ++kx) {
            const int iy = y + ky - 1;
            const int ix = x + kx - 1;
            if (iy >= 0 && iy < HH && ix >= 0 && ix < WW) {
                acc += in[((size_t)(n * HH + iy) * WW + ix) * CCH + c] *
                       dwk[(ky * 3 + kx) * CCH + c];
            }
        }
    }

    __shared__ float s1[CCH];
    __shared__ float s2[CCH];
    s1[c] = acc;
    s2[c] = acc * acc;
    __syncthreads();
    #pragma unroll
    for (int off = 64; off > 0; off >>= 1) {
        if (c < off) { s1[c] += s1[c + off]; s2[c] += s2[c + off]; }
        __syncthreads();
    }
    const float mu  = s1[0] * (1.0f / CCH);
    const float var = s2[0] * (1.0f / CCH) - mu * mu;

    float v = (acc - mu) * rsqrtf(var + 1e-6f) * gamma[c] + beta[c];
    // exact GELU: 0.5*v*(1+erf(v/sqrt(2)))
    v = 0.5f * v * (1.0f + erff(v * 0.70710678118654752440f));
    x1[(size_t)pix * CCH + c] = v;
}

// ---------------------------------------------------------------------------
// Softmax over P=9 per (pixel, group), in place.
// ---------------------------------------------------------------------------
__global__ __launch_bounds__(256) void softmax9(float* __restrict__ m)
{
    const int idx = blockIdx.x * blockDim.x + threadIdx.x;
    if (idx >= NPIX * GG) return;
    const int pix = idx / GG;
    const int g   = idx - pix * GG;
    float* r = m + (size_t)pix * KMSK + g * PP;

    float mx = r[0];
    #pragma unroll
    for (int p = 1; p < PP; ++p) mx = fmaxf(mx, r[p]);
    float e[PP], s = 0.0f;
    #pragma unroll
    for (int p = 0; p < PP; ++p) { e[p] = expf(r[p] - mx); s += e[p]; }
    const float inv = 1.0f / s;
    #pragma unroll
    for (int p = 0; p < PP; ++p) r[p] = e[p] * inv;
}

// ---------------------------------------------------------------------------
// Deformable sampling core. One thread per output element (pixel, channel).
// Padded-coordinate algebra collapses to: px = x + kx + off_x, py = y + ky + off_y
// (padded coords; the zero-pad border is folded into the bounds check).
// p ordering is x-major (meshgrid indexing='ij' over gx,gy): p = kx*3 + ky.
// ---------------------------------------------------------------------------
__device__ __forceinline__ float fetch_pad(const float* __restrict__ xp,
                                           int n, int iy, int ix, int c)
{
    const int uy = iy - 1;   // padded -> unpadded
    const int ux = ix - 1;
    if (uy < 0 || uy >= HH || ux < 0 || ux >= WW) return 0.0f;
    return xp[((size_t)(n * HH + uy) * WW + ux) * CCH + c];
}

__global__ __launch_bounds__(256) void dcn_sample(
    const float* __restrict__ xproj, const float* __restrict__ offs,
    const float* __restrict__ mask, float* __restrict__ out)
{
    const int idx = blockIdx.x * blockDim.x + threadIdx.x;
    if (idx >= NPIX * CCH) return;
    const int c   = idx & (CCH - 1);
    const int pix = idx >> 7;
    const int g   = c >> 5;
    const int x   = pix % WW;
    const int y   = (pix / WW) % HH;
    const int n   = pix / (HH * WW);

    const float* __restrict__ off = offs + (size_t)pix * KOFF + g * (PP * 2);
    const float* __restrict__ mk  = mask + (size_t)pix * KMSK + g * PP;

    float acc = 0.0f;
    #pragma unroll
    for (int p = 0; p < PP; ++p) {
        const int kx = p / 3;
        const int ky = p - kx * 3;
        const float px = (float)(x + kx) + off[2 * p + 0];
        const float py = (float)(y + ky) + off[2 * p + 1];
        const float fx = floorf(px);
        const float fy = floorf(py);
        const int x0 = (int)fx;
        const int y0 = (int)fy;
        const float wx1 = px - fx, wy1 = py - fy;
        const float wx0 = 1.0f - wx1, wy0 = 1.0f - wy1;

        const float v00 = fetch_pad(xproj, n, y0,     x0,     c);
        const float v01 = fetch_pad(xproj, n, y0,     x0 + 1, c);
        const float v10 = fetch_pad(xproj, n, y0 + 1, x0,     c);
        const float v11 = fetch_pad(xproj, n, y0 + 1, x0 + 1, c);

        acc += mk[p] * (v00 * (wy0 * wx0) + v01 * (wy0 * wx1) +
                        v10 * (wy1 * wx0) + v11 * (wy1 * wx1));
    }
    out[idx] = acc;
}

// ---------------------------------------------------------------------------
// Host-side launch
// ---------------------------------------------------------------------------
static inline int gemm_blocks(int M, int N)
{
    const int tilesN = (N + 15) >> 4;
    const int waves  = (M >> 4) * tilesN;
    return (waves + 7) / 8;   // 8 wave32 per 256-thread block
}

extern "C" void kernel_launch(void* const* d_in, const int* in_sizes, int n_in,
                              void* d_out, int out_size, void* d_ws, size_t ws_size,
                              hipStream_t stream)
{
    (void)in_sizes; (void)n_in; (void)out_size; (void)ws_size;

    const float* input    = (const float*)d_in[0];
    const float* w_in     = (const float*)d_in[1];
    const float* b_in     = (const float*)d_in[2];
    const float* w_out    = (const float*)d_in[3];
    const float* b_out    = (const float*)d_in[4];
    const float* w_off    = (const float*)d_in[5];
    const float* b_off    = (const float*)d_in[6];
    const float* w_mask   = (const float*)d_in[7];
    const float* b_mask   = (const float*)d_in[8];
    const float* dw_kern  = (const float*)d_in[9];
    const float* dw_bias  = (const float*)d_in[10];
    const float* ln_gamma = (const float*)d_in[11];
    const float* ln_beta  = (const float*)d_in[12];
    float* out = (float*)d_out;

    // workspace carve-up (floats): xproj | x1 | offsets | mask | sampled
    float* xproj   = (float*)d_ws;
    float* x1      = xproj   + (size_t)NPIX * CCH;
    float* offsb   = x1      + (size_t)NPIX * CCH;
    float* maskb   = offsb   + (size_t)NPIX * KOFF;
    float* sampled = maskb   + (size_t)NPIX * KMSK;

    // 1) input projection: xproj = input @ w_in + b_in
    wmma_gemm_f32_bias<CCH><<<gemm_blocks(NPIX, CCH), 256, 0, stream>>>(
        input, w_in, b_in, xproj, NPIX, CCH);

    // 2) depthwise conv + LayerNorm + exact GELU -> x1
    dw_ln_gelu<<<NPIX, 128, 0, stream>>>(
        input, dw_kern, dw_bias, ln_gamma, ln_beta, x1);

    // 3) offsets: offsb = x1 @ w_off + b_off  (N=72)
    wmma_gemm_f32_bias<CCH><<<gemm_blocks(NPIX, KOFF), 256, 0, stream>>>(
        x1, w_off, b_off, offsb, NPIX, KOFF);

    // 4) mask logits: maskb = x1 @ w_mask + b_mask  (N=36)
    wmma_gemm_f32_bias<CCH><<<gemm_blocks(NPIX, KMSK), 256, 0, stream>>>(
        x1, w_mask, b_mask, maskb, NPIX, KMSK);

    // 5) softmax over P=9 per (pixel, group), in place
    softmax9<<<(NPIX * GG + 255) / 256, 256, 0, stream>>>(maskb);

    // 6) deformable bilinear sampling -> sampled
    dcn_sample<<<(NPIX * CCH + 255) / 256, 256, 0, stream>>>(
        xproj, offsb, maskb, sampled);

    // 7) output projection: out = sampled @ w_out + b_out
    wmma_gemm_f32_bias<CCH><<<gemm_blocks(NPIX, CCH), 256, 0, stream>>>(
        sampled, w_out, b_out, out, NPIX, CCH);
}